// AdaptiveDisLoss_9577777070150
// MI455X (gfx1250) — compile-verified
//
#include <hip/hip_runtime.h>
#include <stdint.h>

// ---------------- problem constants (match reference) ----------------
#define CDIM        81        // NUM_CLASSES + 1
#define NUM_BASE    60
#define TILE_ROWS   64        // rows per TDM tile
#define NB_MAIN     1024      // blocks for the main pass (fixed -> deterministic)

typedef __attribute__((ext_vector_type(4))) unsigned int u32x4;
typedef __attribute__((ext_vector_type(8))) int          i32x8;
typedef __attribute__((ext_vector_type(4))) int          i32x4;

// ---------------- wave32 butterfly reductions ----------------
__device__ __forceinline__ float wave_sum_f(float v) {
#pragma unroll
  for (int m = 16; m > 0; m >>= 1) v += __shfl_xor(v, m, 32);
  return v;
}
__device__ __forceinline__ float wave_max_f(float v) {
#pragma unroll
  for (int m = 16; m > 0; m >>= 1) v = fmaxf(v, __shfl_xor(v, m, 32));
  return v;
}

// ---------------- TDM: async load of a 64x81 f32 tile into LDS ----------------
// Descriptor layout per CDNA5 ISA ch.8 (Tensor DMA Descriptor D#).
// Group0: [1:0]=count(1=valid), [63:32]=lds_addr bytes, [120:64]=global_addr, [127:126]=type(2)
// Group1: data_size=2 (4B), tensor_dim0=81, tensor_dim1=rows remaining,
//         tile_dim0=81, tile_dim1=64, tensor_dim0_stride=81 elements.
__device__ __forceinline__ void tdm_load_tile(const float* base, unsigned ldsByteOff,
                                              long long rowStart, unsigned rowsRemaining) {
  unsigned long long ga = (unsigned long long)(uintptr_t)base
                        + (unsigned long long)rowStart * (unsigned long long)(CDIM * 4);
  u32x4 g0;
  g0[0] = 1u;                                            // count=1 (valid), user mode
  g0[1] = ldsByteOff;                                    // lds_addr (bytes)
  g0[2] = (unsigned)(ga & 0xFFFFFFFFull);                // global_addr[31:0]
  g0[3] = (unsigned)((ga >> 32) & 0x1FFFFFFull)          // global_addr[56:32]
        | (2u << 30);                                    // type = 2 ("image")
  i32x8 g1;
  g1[0] = (int)(2u << 16);                               // workgroup_mask=0, data_size=2 (4B)
  g1[1] = (int)(((unsigned)CDIM & 0xFFFFu) << 16);       // tensor_dim0[15:0] at bits 63:48
  g1[2] = (int)((rowsRemaining & 0xFFFFu) << 16);        // tensor_dim0 hi=0 | tensor_dim1[15:0]
  g1[3] = (int)(((rowsRemaining >> 16) & 0xFFFFu)        // tensor_dim1[31:16]
        | (((unsigned)CDIM & 0xFFFFu) << 16));           // tile_dim0 = 81
  g1[4] = TILE_ROWS;                                     // tile_dim1 = 64, tile_dim2 = 0
  g1[5] = CDIM;                                          // tensor_dim0_stride[31:0] = 81
  g1[6] = 0;                                             // stride0 hi | stride1 lo
  g1[7] = 0;                                             // stride1 hi
  i32x4 gz4 = {0, 0, 0, 0};                              // groups 2/3 unused (2-D tensor)
  i32x8 gz8 = {0, 0, 0, 0, 0, 0, 0, 0};                  // extra group (clang-23 6-arg form)
  __builtin_amdgcn_tensor_load_to_lds(g0, g1, gz4, gz4, gz8, 0);
}

// ---------------- kernels ----------------
__global__ void adis_zero(unsigned* counts) {
  if (threadIdx.x < 128) counts[threadIdx.x] = 0u;
}

__global__ void adis_count(const int* __restrict__ labels, const float* __restrict__ lw,
                           int n, unsigned* __restrict__ counts) {
  int i = blockIdx.x * blockDim.x + threadIdx.x;
  int stride = gridDim.x * blockDim.x;
  for (; i < n; i += stride)
    if (lw[i] > 0.0f) atomicAdd(&counts[labels[i]], 1u);
}

struct __align__(16) SMem {
  float    tile[2][TILE_ROWS * CDIM];   // double-buffered TDM target (offset 0 in LDS)
  double   wsum[8][3];
  unsigned wcnt[8][3];
};

__global__ void __launch_bounds__(256)
adis_main(const float* __restrict__ cls, const int* __restrict__ labels,
          const float* __restrict__ lw, const unsigned* __restrict__ counts,
          int N, double* __restrict__ psum, unsigned* __restrict__ pcnt) {
  __shared__ SMem sm;
  const int tid  = threadIdx.x;
  const int wid  = tid >> 5;
  const int lane = tid & 31;

  double   gsum[3] = {0.0, 0.0, 0.0};
  unsigned gcnt[3] = {0u, 0u, 0u};

  const int nTiles = (N + TILE_ROWS - 1) / TILE_ROWS;

  int t = blockIdx.x;
  if (t < nTiles && wid == 0) {
    long long rs = (long long)t * TILE_ROWS;
    tdm_load_tile(cls, 0u, rs, (unsigned)(N - rs));
  }
  int buf = 0;
  for (; t < nTiles; t += gridDim.x) {
    if (wid == 0) __builtin_amdgcn_s_wait_tensorcnt(0);   // tile t landed in LDS
    __syncthreads();                                      // visible to all 8 waves
    const int tn = t + (int)gridDim.x;
    if (tn < nTiles && wid == 0) {                        // prefetch tile t+stride
      long long rs = (long long)tn * TILE_ROWS;
      tdm_load_tile(cls, (unsigned)((buf ^ 1) * TILE_ROWS * CDIM * 4), rs, (unsigned)(N - rs));
    }
    const float* tb = sm.tile[buf];
#pragma unroll 1
    for (int rr = 0; rr < 8; ++rr) {                      // each wave: 8 rows of the tile
      const int rloc = wid * 8 + rr;
      const long long grow = (long long)t * TILE_ROWS + rloc;
      if (grow >= N) continue;                            // wave-uniform
      const float* rp = tb + rloc * CDIM;
      // 81 columns in 3 regs per lane
      float x0 = rp[lane];
      float x1 = rp[32 + lane];
      float x2 = (lane < CDIM - 64) ? rp[64 + lane] : -3.0e38f;
      float mx = wave_max_f(fmaxf(x0, fmaxf(x1, x2)));
      float e0 = __expf(x0 - mx);
      float e1 = __expf(x1 - mx);
      float e2 = (lane < CDIM - 64) ? __expf(x2 - mx) : 0.0f;
      float denom = wave_sum_f(e0 + e1 + e2);
      const int   lbl = labels[grow];
      const float w   = lw[grow];
      const int k  = lbl >> 5;
      const int lt = lbl & 31;
      float esel  = (k == 0) ? e0 : ((k == 1) ? e1 : e2);
      float etrue = __shfl(esel, lt, 32);
      float inv = 1.0f / denom;
      float pt  = etrue * inv;
      // per-column margin terms, excluding the true-class column
      float acc = 0.0f;
      {
        float d = fminf(fmaxf(pt - e0 * inv, 1.0e-3f), 1.0f);
        float g = fminf(-__logf(d), 5.0f);
        if (lane != lbl) acc += g;
      }
      {
        float d = fminf(fmaxf(pt - e1 * inv, 1.0e-3f), 1.0f);
        float g = fminf(-__logf(d), 5.0f);
        if (32 + lane != lbl) acc += g;
      }
      if (lane < CDIM - 64) {
        float d = fminf(fmaxf(pt - e2 * inv, 1.0e-3f), 1.0f);
        float g = fminf(-__logf(d), 5.0f);
        if (64 + lane != lbl) acc += g;
      }
      float S  = wave_sum_f(acc);
      float om = fminf(fmaxf(1.0f - pt, 1.0e-4f), 1.0f);
      float rowsum = om * om * S;
      // group selection: base / novel (with count>=2 gate) / neg
      bool valid = w > 0.0f;
      int  grp   = (lbl < NUM_BASE) ? 0 : ((lbl < CDIM - 1) ? 1 : 2);
      bool enough = (grp == 2) ? true : (counts[lbl] >= 2u);
      if (valid && enough) { gsum[grp] += (double)rowsum; gcnt[grp] += 1u; }
    }
    __syncthreads();                                      // everyone done with buf -> reusable
    buf ^= 1;
  }

  if (lane == 0) {
#pragma unroll
    for (int g = 0; g < 3; ++g) { sm.wsum[wid][g] = gsum[g]; sm.wcnt[wid][g] = gcnt[g]; }
  }
  __syncthreads();
  if (tid == 0) {                                         // fixed-order per-block combine
#pragma unroll
    for (int g = 0; g < 3; ++g) {
      double s = 0.0; unsigned n = 0;
      for (int wv = 0; wv < 8; ++wv) { s += sm.wsum[wv][g]; n += sm.wcnt[wv][g]; }
      psum[blockIdx.x * 3 + g] = s;
      pcnt[blockIdx.x * 3 + g] = n;
    }
  }
}

__global__ void __launch_bounds__(256)
adis_final(const double* __restrict__ psum, const unsigned* __restrict__ pcnt,
           int nb, float* __restrict__ out) {
  __shared__ double   ss[256][3];
  __shared__ unsigned sn[256][3];
  const int tid = threadIdx.x;
  double s[3] = {0.0, 0.0, 0.0}; unsigned n[3] = {0u, 0u, 0u};
  for (int i = tid; i < nb; i += 256)
#pragma unroll
    for (int g = 0; g < 3; ++g) { s[g] += psum[i * 3 + g]; n[g] += pcnt[i * 3 + g]; }
#pragma unroll
  for (int g = 0; g < 3; ++g) { ss[tid][g] = s[g]; sn[tid][g] = n[g]; }
  __syncthreads();
  for (int off = 128; off > 0; off >>= 1) {
    if (tid < off)
#pragma unroll
      for (int g = 0; g < 3; ++g) { ss[tid][g] += ss[tid + off][g]; sn[tid][g] += sn[tid + off][g]; }
    __syncthreads();
  }
  if (tid == 0) {
    const float W[3] = {1.0f / 30.0f, 0.1f, 0.001f};      // W_BASE, W_NOVEL, W_NEG
#pragma unroll
    for (int g = 0; g < 3; ++g) {
      double  sg = ss[0][g];
      unsigned ng = sn[0][g];
      double mean = (ng > 0u) ? sg / ((double)ng * (double)(CDIM - 1)) : 0.0;
      out[g] = fminf((float)mean * W[g], 1.0f);           // min(mean*W, MAX_LOSS)
    }
  }
}

// ---------------- host entry ----------------
extern "C" void kernel_launch(void* const* d_in, const int* in_sizes, int n_in,
                              void* d_out, int out_size, void* d_ws, size_t ws_size,
                              hipStream_t stream) {
  const float* cls    = (const float*)d_in[0];
  const int*   labels = (const int*)d_in[1];
  const float* lwts   = (const float*)d_in[2];
  const int N = in_sizes[1];
  float* out = (float*)d_out;

  // workspace layout: [0,512): class counts (81 ints, padded)
  //                   [512, 512+NB*3*8): per-block double partial sums
  //                   then NB*3*4 bytes of per-block counts
  unsigned* counts = (unsigned*)d_ws;
  double*   psum   = (double*)((char*)d_ws + 512);
  unsigned* pcnt   = (unsigned*)((char*)d_ws + 512 + (size_t)NB_MAIN * 3 * sizeof(double));

  adis_zero<<<1, 128, 0, stream>>>(counts);
  adis_count<<<512, 256, 0, stream>>>(labels, lwts, N, counts);
  adis_main<<<NB_MAIN, 256, 0, stream>>>(cls, labels, lwts, counts, N, psum, pcnt);
  adis_final<<<1, 256, 0, stream>>>(psum, pcnt, NB_MAIN, out);
}